// LinearSpatialAttention_78331613545272
// MI455X (gfx1250) — compile-verified
//
#include <hip/hip_runtime.h>
#include <hip/hip_bf16.h>

typedef __attribute__((ext_vector_type(2))) float v2f;
typedef __attribute__((ext_vector_type(8))) float v8f;

namespace {
constexpr int Bb  = 16;
constexpr int Cc  = 64;
constexpr int Nn  = 256 * 256;   // H*W
constexpr int DIM = 8;
constexpr float EPS = 1e-6f;

constexpr int PIX_PER_WAVE   = 256;  // 16 tiles of 16 pixels per wave
constexpr int WAVES_PER_BLK  = 8;    // 256 threads
constexpr int WAVES_PER_BATCH = Nn / PIX_PER_WAVE;           // 256
constexpr int GRID_BLOCKS    = Bb * WAVES_PER_BATCH / WAVES_PER_BLK; // 512
}

// ---------------------------------------------------------------------------
// zero the kv accumulator (16 batches x 9 x 8 floats) in workspace
// ---------------------------------------------------------------------------
__global__ void kv_init_kernel(float* __restrict__ kv) {
    int i = blockIdx.x * blockDim.x + threadIdx.x;
    if (i < Bb * 72) kv[i] = 0.f;
}

// ---------------------------------------------------------------------------
// Pass 1: kv[b,d,e] = sum_n v_pad[b,d,n] * relu(k)[b,e,n]
// k;v = (qkv_w rows 8..23) @ x  via chained V_WMMA_F32_16X16X4_F32 (K=64)
// ---------------------------------------------------------------------------
__global__ void __launch_bounds__(256)
pass1_kernel(const float* __restrict__ x,
             const float* __restrict__ qkv_w,
             float* __restrict__ kv)
{
    const int lane = threadIdx.x & 31;
    const int wave = blockIdx.x * WAVES_PER_BLK + (threadIdx.x >> 5);
    const int b    = wave / WAVES_PER_BATCH;
    const int n0b  = (wave % WAVES_PER_BATCH) * PIX_PER_WAVE;
    const int half = lane >> 4;     // 0: K=4k..4k+1 side, 1: K=4k+2..4k+3 side
    const int lan  = lane & 15;

    // A operand = [Wk ; Wv] = qkv_w rows 8..23 (16x64), preloaded for 16 K-steps
    v2f a[16];
    {
        const float* wr = qkv_w + (8 + lan) * Cc;
        #pragma unroll
        for (int ks = 0; ks < 16; ks++) {
            const int kb = ks * 4 + half * 2;
            a[ks].x = wr[kb + 0];
            a[ks].y = wr[kb + 1];
        }
    }

    float acc[40];                              // low half: v rows 0..3 + ones row
    #pragma unroll                              // high half: v rows 4..7
    for (int i = 0; i < 40; i++) acc[i] = 0.f;

    const float* xb = x + (size_t)b * Cc * Nn;

    for (int t = 0; t < PIX_PER_WAVE / 16; t++) {
        const int n0 = n0b + t * 16;

        // qkv GEMM tile: D(16x16) over K=64 in 16 WMMA steps
        v8f d = {};
        #pragma unroll
        for (int ks = 0; ks < 16; ks++) {
            const int c0 = ks * 4 + half * 2;
            v2f bop;
            bop.x = xb[(size_t)(c0 + 0) * Nn + n0 + lan];
            bop.y = xb[(size_t)(c0 + 1) * Nn + n0 + lan];
            d = __builtin_amdgcn_wmma_f32_16x16x4_f32(
                    false, a[ks], false, bop, (short)0, d, false, false);
        }

        // D rows 0..7 = k_pre (low lanes), rows 8..15 = v (high lanes).
        // Pair k and v of the same pixel across wave halves.
        float kq[8], vv[8];
        #pragma unroll
        for (int j = 0; j < 8; j++) {
            const float mine  = d[j];
            const float other = __shfl_xor(mine, 16, 32);
            const float kpre  = half ? other : mine;
            kq[j] = fmaxf(kpre, 0.f);
            vv[j] = half ? mine : other;
        }

        // outer-product accumulate (each pixel counted once across the halves)
        #pragma unroll
        for (int dd = 0; dd < 4; dd++) {
            const float vval = half ? vv[dd + 4] : vv[dd];
            #pragma unroll
            for (int e = 0; e < 8; e++)
                acc[dd * 8 + e] = fmaf(vval, kq[e], acc[dd * 8 + e]);
        }
        #pragma unroll
        for (int e = 0; e < 8; e++)             // ones row (v_pad row 8), low half only
            acc[32 + e] += half ? 0.f : kq[e];
    }

    // butterfly reduce within each 16-lane half
    #pragma unroll
    for (int i = 0; i < 40; i++) {
        float v = acc[i];
        v += __shfl_xor(v, 1, 32);
        v += __shfl_xor(v, 2, 32);
        v += __shfl_xor(v, 4, 32);
        v += __shfl_xor(v, 8, 32);
        acc[i] = v;
    }

    float* kvb = kv + b * 72;
    if (lane == 0) {
        #pragma unroll
        for (int dd = 0; dd < 4; dd++)
            for (int e = 0; e < 8; e++)
                atomicAdd(&kvb[dd * 8 + e], acc[dd * 8 + e]);
        #pragma unroll
        for (int e = 0; e < 8; e++)
            atomicAdd(&kvb[64 + e], acc[32 + e]);
    }
    if (lane == 16) {
        #pragma unroll
        for (int dd = 0; dd < 4; dd++)
            for (int e = 0; e < 8; e++)
                atomicAdd(&kvb[(dd + 4) * 8 + e], acc[dd * 8 + e]);
    }
}

// ---------------------------------------------------------------------------
// Pass 2: q = relu(Wq x); s = kv q; t = s[0:8]/(s[8]+eps);
//         out = proj_w t + x     (proj via 4 WMMA M-tiles, residual fused in C)
// ---------------------------------------------------------------------------
__global__ void __launch_bounds__(256)
pass2_kernel(const float* __restrict__ x,
             const float* __restrict__ qkv_w,
             const float* __restrict__ proj_w,
             const float* __restrict__ kv,
             float* __restrict__ out)
{
    const int lane = threadIdx.x & 31;
    const int wave = blockIdx.x * WAVES_PER_BLK + (threadIdx.x >> 5);
    const int b    = wave / WAVES_PER_BATCH;
    const int n0b  = (wave % WAVES_PER_BATCH) * PIX_PER_WAVE;
    const int half = lane >> 4;
    const int lan  = lane & 15;

    // A operand for q: Wq rows 0..7, zero-padded rows 8..15
    v2f aq[16];
    {
        const float* wr = qkv_w + lan * Cc;     // valid only for lan < 8
        #pragma unroll
        for (int ks = 0; ks < 16; ks++) {
            const int kb = ks * 4 + half * 2;
            aq[ks].x = (lan < 8) ? wr[kb + 0] : 0.f;
            aq[ks].y = (lan < 8) ? wr[kb + 1] : 0.f;
        }
    }

    // A operand for proj: 4 tiles of 16 rows, K=8 in 2 steps
    v2f ap[4][2];
    #pragma unroll
    for (int mb = 0; mb < 4; mb++) {
        const float* pr = proj_w + (size_t)(mb * 16 + lan) * DIM;
        #pragma unroll
        for (int ks = 0; ks < 2; ks++) {
            const int kb = ks * 4 + half * 2;
            ap[mb][ks].x = pr[kb + 0];
            ap[mb][ks].y = pr[kb + 1];
        }
    }

    // kv matrix for this batch (uniform per block; 72 floats)
    float kvr[72];
    #pragma unroll
    for (int i = 0; i < 72; i++) kvr[i] = kv[b * 72 + i];

    const float* xb = x   + (size_t)b * Cc * Nn;
    float*       ob = out + (size_t)b * Cc * Nn;

    for (int t = 0; t < PIX_PER_WAVE / 16; t++) {
        const int n0 = n0b + t * 16;

        // q = relu(Wq @ x_tile): D rows 0..7 valid, land fully in low lanes
        v8f dq = {};
        #pragma unroll
        for (int ks = 0; ks < 16; ks++) {
            const int c0 = ks * 4 + half * 2;
            v2f bop;
            bop.x = xb[(size_t)(c0 + 0) * Nn + n0 + lan];
            bop.y = xb[(size_t)(c0 + 1) * Nn + n0 + lan];
            dq = __builtin_amdgcn_wmma_f32_16x16x4_f32(
                     false, aq[ks], false, bop, (short)0, dq, false, false);
        }

        float q[8];
        #pragma unroll
        for (int j = 0; j < 8; j++) q[j] = fmaxf(dq[j], 0.f);

        // s = kv @ q, normalize by denominator row (meaningful in low lanes)
        float s8 = 0.f;
        #pragma unroll
        for (int e = 0; e < 8; e++) s8 = fmaf(kvr[64 + e], q[e], s8);
        const float inv = 1.f / (s8 + EPS);

        float tt[8];
        #pragma unroll
        for (int dd = 0; dd < 8; dd++) {
            float s = 0.f;
            #pragma unroll
            for (int e = 0; e < 8; e++) s = fmaf(kvr[dd * 8 + e], q[e], s);
            tt[dd] = s * inv;
        }

        // rebuild B layout for proj WMMA: high half needs K=+2,+3 from partner
        const float u2 = __shfl_xor(tt[2], 16, 32);
        const float u3 = __shfl_xor(tt[3], 16, 32);
        const float u6 = __shfl_xor(tt[6], 16, 32);
        const float u7 = __shfl_xor(tt[7], 16, 32);
        v2f b0, b1;
        b0.x = half ? u2 : tt[0];
        b0.y = half ? u3 : tt[1];
        b1.x = half ? u6 : tt[4];
        b1.y = half ? u7 : tt[5];

        // out tile = proj_w @ t + x  (residual pre-loaded into C in D-layout)
        #pragma unroll
        for (int mb = 0; mb < 4; mb++) {
            v8f cacc;
            #pragma unroll
            for (int j = 0; j < 8; j++) {
                const int c = mb * 16 + half * 8 + j;
                cacc[j] = xb[(size_t)c * Nn + n0 + lan];
            }
            cacc = __builtin_amdgcn_wmma_f32_16x16x4_f32(
                       false, ap[mb][0], false, b0, (short)0, cacc, false, false);
            cacc = __builtin_amdgcn_wmma_f32_16x16x4_f32(
                       false, ap[mb][1], false, b1, (short)0, cacc, false, false);
            #pragma unroll
            for (int j = 0; j < 8; j++) {
                const int c = mb * 16 + half * 8 + j;
                ob[(size_t)c * Nn + n0 + lan] = cacc[j];
            }
        }
    }
}

// ---------------------------------------------------------------------------
extern "C" void kernel_launch(void* const* d_in, const int* in_sizes, int n_in,
                              void* d_out, int out_size, void* d_ws, size_t ws_size,
                              hipStream_t stream) {
    (void)in_sizes; (void)n_in; (void)out_size; (void)ws_size;
    const float* x      = (const float*)d_in[0];
    const float* qkv_w  = (const float*)d_in[1];
    const float* proj_w = (const float*)d_in[2];
    float*       out    = (float*)d_out;
    float*       kvws   = (float*)d_ws;          // 16*72 floats

    kv_init_kernel<<<dim3((Bb * 72 + 255) / 256), dim3(256), 0, stream>>>(kvws);
    pass1_kernel<<<dim3(GRID_BLOCKS), dim3(256), 0, stream>>>(x, qkv_w, kvws);
    pass2_kernel<<<dim3(GRID_BLOCKS), dim3(256), 0, stream>>>(x, qkv_w, proj_w, kvws, out);
}